// BatchedFerroelectricBasis_87978110091297
// MI455X (gfx1250) — compile-verified
//
#include <hip/hip_runtime.h>
#include <hip/hip_bf16.h>

typedef __attribute__((ext_vector_type(2))) float v2f;
typedef __attribute__((ext_vector_type(8))) float v8f;

#define B_  512
#define I_  128
#define O_  64
#define NB_ 8
#define LOG2E 1.4426950408889634f
#define CG  (10.0f * LOG2E)   // exp(10*t) = exp2(CG*t)
#define KT  (-2.0f * LOG2E)   // tanh(z) = 2/(1+exp2(KT*z)) - 1

// ---------- prep 1: per-(b,i) values, transposed to [I,B] for coalescing ----
// pxu[i*B + b] = { x, exp(10x), -0.4 * sigmoid(-10*(x-prev)), 0 }
__global__ __launch_bounds__(256) void prep_bi(const float* __restrict__ x,
                                               float4* __restrict__ pxu,
                                               float* __restrict__ bsum) {
  int tid = blockIdx.x * 256 + threadIdx.x;   // 65536 threads: tid = i*512 + b
  if (tid < O_) bsum[tid] = 0.0f;             // re-zeroed every launch (deterministic)
  int b = tid & (B_ - 1);
  int i = tid >> 9;
  float xv = x[b * I_ + i];
  float pv = (b == 0) ? 0.0f : x[(b - 1) * I_ + i];
  float u  = __builtin_amdgcn_exp2f(CG * xv);
  float g  = __builtin_amdgcn_rcpf(1.0f + __builtin_amdgcn_exp2f(CG * (xv - pv)));
  pxu[tid] = make_float4(xv, u, -0.4f * g, 0.0f);
}

// ---------- prep 2: per-(i,o,n) table + bias/coef reduction ------------------
// tab[(o*I + i)*NB + n] = { KT*k, exp(10*Ec), KT*k*Ec, 2*Ps*coef }
// bsum[o] += bias*coef - Ps*coef   (folds the "-1" of tanh rewrite)
__global__ __launch_bounds__(256) void prep_tab(const float* __restrict__ k,
                                                const float* __restrict__ Ec,
                                                const float* __restrict__ Ps,
                                                const float* __restrict__ bias,
                                                const float* __restrict__ coef,
                                                float4* __restrict__ tab,
                                                float* __restrict__ bsum) {
  int tid = blockIdx.x * 256 + threadIdx.x;   // 65536 = I*O*NB, input layout (i,o,n)
  int n = tid & (NB_ - 1);
  int o = (tid >> 3) & (O_ - 1);
  int i = tid >> 9;
  float kv = k[tid], ec = Ec[tid], ps = Ps[tid], bi = bias[tid], cf = coef[tid];
  float k2 = KT * kv;
  float Ep = __builtin_amdgcn_exp2f(CG * ec);
  tab[(o * I_ + i) * NB_ + n] = make_float4(k2, Ep, k2 * ec, 2.0f * ps * cf);
  atomicAdd(&bsum[o], bi * cf - ps * cf);
}

// ---------- main: one wave per (16-batch tile, o); WMMA 4-way reduction ------
__global__ __launch_bounds__(256) void ferro_main(const float4* __restrict__ pxu,
                                                  const float4* __restrict__ tab,
                                                  const float* __restrict__ bsum,
                                                  float* __restrict__ out) {
  int lane = threadIdx.x & 31;
  int wave = (blockIdx.x * 256 + threadIdx.x) >> 5;  // 0..2047
  int o  = wave & (O_ - 1);
  int bt = wave >> 6;                                // 0..31 batch tiles
  int h  = lane >> 4;                                // which K half (A layout)
  int r  = lane & 15;                                // batch row in tile (A row M)
  int b  = bt * 16 + r;

  const float4* tabo = tab + (size_t)o * (I_ * NB_);

  float acc0 = 0.0f, acc1 = 0.0f;
  #pragma unroll
  for (int cc = 0; cc < 2; ++cc) {
    int c = 2 * h + cc;                 // K-chunk 0..3, 256 K-values each
    float a = 0.0f;
    for (int il = 0; il < 32; ++il) {
      int i = c * 32 + il;
      float4 p  = pxu[i * B_ + b];      // coalesced: lanes 0-15 consecutive b
      float xv = p.x, uv = p.y, gh = p.z;
      const float4* t8 = tabo + i * NB_;
      #pragma unroll
      for (int n = 0; n < NB_; ++n) {
        float4 t  = t8[n];              // {k2, Ep, kEc2, Pc2} — 128b load, same address per 16-lane half
        float k2x = t.x * xv;
        float cn  = __builtin_amdgcn_rcpf(fmaf(t.y, uv, 1.0f));   // crossed_neg
        float bm  = fmaf(gh, cn, 1.0f);                           // branch momentum
        float z   = fmaf(t.z, bm, k2x);                           // KT * k*(x+Ec*bm)
        float e   = __builtin_amdgcn_exp2f(z);
        float rr  = __builtin_amdgcn_rcpf(e + 1.0f);              // (tanh+1)/2
        a = fmaf(t.w, rr, a);                                     // += 2*Ps*coef * rr
      }
    }
    if (cc == 0) acc0 = a; else acc1 = a;
  }

  // A[16x4] rows = batch rows, K = 4 partial chunks; B = ones(4x16)
  // D[m, *] = row sum -> full contraction for each batch row. EXEC is all 1s here.
  v2f A;  A.x = acc0; A.y = acc1;
  v2f Bv; Bv.x = 1.0f; Bv.y = 1.0f;
  v8f C = {};
  v8f D = __builtin_amdgcn_wmma_f32_16x16x4_f32(
      /*neg_a=*/false, A, /*neg_b=*/false, Bv,
      /*c_mod=*/(short)0, C, /*reuse_a=*/false, /*reuse_b=*/false);

  if (r == 0) {                         // lane 0 -> rows 0..7, lane 16 -> rows 8..15
    float bs = bsum[o];
    #pragma unroll
    for (int v = 0; v < 8; ++v)
      out[(bt * 16 + v + 8 * h) * O_ + o] = D[v] + bs;
  }
}

extern "C" void kernel_launch(void* const* d_in, const int* in_sizes, int n_in,
                              void* d_out, int out_size, void* d_ws, size_t ws_size,
                              hipStream_t stream) {
  const float* x    = (const float*)d_in[0];
  const float* k    = (const float*)d_in[1];
  const float* Ec   = (const float*)d_in[2];
  const float* Ps   = (const float*)d_in[3];
  const float* bias = (const float*)d_in[4];
  const float* coef = (const float*)d_in[5];

  char* ws = (char*)d_ws;
  float4* pxu  = (float4*)(ws);                 // 65536 * 16B = 1 MiB
  float4* tab  = (float4*)(ws + (1u << 20));    // 65536 * 16B = 1 MiB
  float*  bsum = (float*)(ws + (2u << 20));     // 64 * 4B

  prep_bi <<<256, 256, 0, stream>>>(x, pxu, bsum);
  prep_tab<<<256, 256, 0, stream>>>(k, Ec, Ps, bias, coef, tab, bsum);
  ferro_main<<<256, 256, 0, stream>>>(pxu, tab, bsum, (float*)d_out);
}